// NCykParser_34144990003767
// MI455X (gfx1250) — compile-verified
//
#include <hip/hip_runtime.h>
#include <stdint.h>

#define RULES 64
#define NTOK  16
#define CHART_FLOATS ((NTOK + 1) * NTOK * RULES)   // 17*16*64 floats = 68KB

typedef unsigned int u32x4 __attribute__((ext_vector_type(4)));
typedef int          i32x8 __attribute__((ext_vector_type(8)));
typedef int          i32x4 __attribute__((ext_vector_type(4)));

// Dynamic-LDS layout (float indices). One shared allocation starting at LDS offset 0.
#define UOFF 0
#define VOFF 64
#define TOFF 128
#define TILE_FLOATS (64 * 256)                 // full tile: 64 'a' rows x (4 b x 64 c)
#define LDS_FLOATS  (TOFF + 2 * TILE_FLOATS)   // U + V + double-buffered tiles
#define LDS_BYTES   (LDS_FLOATS * 4)           // 131,584 B  (< 320KB WGP LDS)

// ---------------------------------------------------------------------------
// TDM: per-wave DMA of an 8-row W slice, W[a0..a0+8, b0..b0+4, 0..64), into
// this wave's private LDS tile region. 2D descriptor per CDNA5 ISA sec 8.3/8.4.
//   x (dim0): 256 contiguous elements (4 b-rows x 64 c)
//   y (dim1): 8 'a' rows, stride 4096 elements
// Every wave issues exactly its own slice (uniform execution, no EXEC games);
// TDM ops from one wave complete in-order, so TENSORcnt<=1 pipelining is exact.
// ---------------------------------------------------------------------------
__device__ __forceinline__ void tdm_issue_tile(const float* W, int wv, int t, int buf) {
  const uint64_t ga = (uint64_t)(uintptr_t)W +
      ((uint64_t)(wv * 8) * (RULES * RULES) + (uint64_t)(t * 4 * RULES)) * 4ull;
  const uint32_t laddr = (uint32_t)(TOFF + buf * TILE_FLOATS + wv * (8 * 256)) * 4u;

  u32x4 g0;
  g0.x = 1u;                                                  // count=1, user desc
  g0.y = laddr;                                               // lds_addr (bytes)
  g0.z = (uint32_t)ga;                                        // global_addr[31:0]
  g0.w = (uint32_t)((ga >> 32) & 0x01FFFFFFu) | (2u << 30);   // addr[56:32] | type=2

  i32x8 g1;
  g1[0] = (int)(2u << 16);     // workgroup_mask=0, data_size=2 (4 bytes)
  g1[1] = (int)(4096u << 16);  // tensor_dim0 = 4096 (elements per 'a' row)
  g1[2] = (int)(64u << 16);    // tensor_dim1 = 64
  g1[3] = (int)(256u << 16);   // tile_dim0 = 256
  g1[4] = 8;                   // tile_dim1 = 8, tile_dim2 = 0
  g1[5] = 4096;                // tensor_dim0_stride = 4096 elements
  g1[6] = 0;                   // stride0[47:32]=0, stride1[15:0]=0
  g1[7] = 0;                   // stride1[47:16]=0

  i32x4 gz;
  gz[0] = 0; gz[1] = 0; gz[2] = 0; gz[3] = 0;                 // groups 2/3 unused (2D)

  i32x8 gz8;
  gz8[0] = 0; gz8[1] = 0; gz8[2] = 0; gz8[3] = 0;             // unused trailing group
  gz8[4] = 0; gz8[5] = 0; gz8[6] = 0; gz8[7] = 0;

  // clang-23 / therock lane: 6-arg form (g0, g1, g2, g3, extra, cpol)
  __builtin_amdgcn_tensor_load_to_lds(g0, g1, gz, gz, gz8, 0);
}

// ---------------------------------------------------------------------------
// One (s,k) cell per block. 256 threads: thread = (a = tid>>2, c-quarter = tid&3).
// Each wave streams its own 8 'a'-rows of W through double-buffered LDS tiles
// via TDM, synchronized with s_wait_tensorcnt (in-order per-wave completion).
// ---------------------------------------------------------------------------
__global__ __launch_bounds__(256) void cyk_level_kernel(const float* __restrict__ W,
                                                        float* __restrict__ chart,
                                                        int L) {
  extern __shared__ float lds[];
  const int tid = threadIdx.x;
  const int wv  = __builtin_amdgcn_readfirstlane(tid >> 5);   // wave id (uniform)

  // Kick off the first DMA immediately: it fills this wave's tile region while
  // we stage U/V and cross the barrier.
  tdm_issue_tile(W, wv, 0, 0);

  const int nk = L - 1;
  const int s  = blockIdx.x / nk;
  const int k  = 1 + (blockIdx.x - s * nk);

  // Stage U = chart[k, s, :], V = chart[L-k, s+k, :] into LDS.
  if (tid < 2 * RULES) {
    const int which = tid >> 6;                 // 0 -> U, 1 -> V
    const int r     = tid & (RULES - 1);
    const int lvl   = which ? (L - k) : k;
    const int st    = which ? (s + k) : s;
    lds[(which ? VOFF : UOFF) + r] = chart[(lvl * NTOK + st) * RULES + r];
  }
  __syncthreads();

  const int arow = (tid >> 2) << 8;             // (tid>>2) * 256
  const int c0   = (tid & 3) << 4;              // c-quarter base

  // This thread's 16 V values, held in registers.
  const float4 v0 = *(const float4*)&lds[VOFF + c0 + 0];
  const float4 v1 = *(const float4*)&lds[VOFF + c0 + 4];
  const float4 v2 = *(const float4*)&lds[VOFF + c0 + 8];
  const float4 v3 = *(const float4*)&lds[VOFF + c0 + 12];

  float m = 0.0f;   // starting at 0 == built-in relu of the running max

  for (int t = 0; t < 16; ++t) {
    if (t < 15) {
      tdm_issue_tile(W, wv, t + 1, (t + 1) & 1);   // prefetch next tile
      __builtin_amdgcn_s_wait_tensorcnt(1);        // tile t complete (in-order)
    } else {
      __builtin_amdgcn_s_wait_tensorcnt(0);
    }
    __asm__ __volatile__("" ::: "memory");         // keep LDS reads after the wait

    const float* tp = lds + TOFF + ((t & 1) ? TILE_FLOATS : 0) + arow;
    const float4 ub4 = *(const float4*)&lds[UOFF + t * 4];   // U[4t..4t+4), broadcast
    const float ubs[4] = {ub4.x, ub4.y, ub4.z, ub4.w};
#pragma unroll
    for (int bl = 0; bl < 4; ++bl) {
      const float ub = ubs[bl];
      const float4 w0 = *(const float4*)(tp + bl * 64 + c0 + 0);
      const float4 w1 = *(const float4*)(tp + bl * 64 + c0 + 4);
      const float4 w2 = *(const float4*)(tp + bl * 64 + c0 + 8);
      const float4 w3 = *(const float4*)(tp + bl * 64 + c0 + 12);
      m = fmaxf(m, w0.x * (ub * v0.x)); m = fmaxf(m, w0.y * (ub * v0.y));
      m = fmaxf(m, w0.z * (ub * v0.z)); m = fmaxf(m, w0.w * (ub * v0.w));
      m = fmaxf(m, w1.x * (ub * v1.x)); m = fmaxf(m, w1.y * (ub * v1.y));
      m = fmaxf(m, w1.z * (ub * v1.z)); m = fmaxf(m, w1.w * (ub * v1.w));
      m = fmaxf(m, w2.x * (ub * v2.x)); m = fmaxf(m, w2.y * (ub * v2.y));
      m = fmaxf(m, w2.z * (ub * v2.z)); m = fmaxf(m, w2.w * (ub * v2.w));
      m = fmaxf(m, w3.x * (ub * v3.x)); m = fmaxf(m, w3.y * (ub * v3.y));
      m = fmaxf(m, w3.z * (ub * v3.z)); m = fmaxf(m, w3.w * (ub * v3.w));
    }
    __asm__ __volatile__("" ::: "memory");
  }

  // Reduce the 4 c-quarter lanes (same 'a', adjacent lanes in one wave32).
  m = fmaxf(m, __shfl_xor(m, 1, 32));
  m = fmaxf(m, __shfl_xor(m, 2, 32));

  // Combine across k (and blocks) with a bitwise uint max: all values >= +0.0,
  // so IEEE order == unsigned-integer order. Chart was pre-zeroed.
  if ((tid & 3) == 0) {
    const int a = tid >> 2;
    atomicMax((unsigned int*)&chart[(L * NTOK + s) * RULES + a], __float_as_uint(m));
  }
}

// ---------------------------------------------------------------------------
// Init: zero the whole chart (atomicMax targets must start at +0.0), then
// chart[1, s, :] = E[tokens[s], :].
// ---------------------------------------------------------------------------
__global__ void cyk_init_kernel(const int* __restrict__ tokens,
                                const float* __restrict__ E,
                                float* __restrict__ chart) {
  const int tid = threadIdx.x;
  for (int i = tid; i < CHART_FLOATS; i += blockDim.x) chart[i] = 0.0f;
  __syncthreads();
  for (int i = tid; i < NTOK * RULES; i += blockDim.x) {
    const int s = i >> 6;
    const int r = i & (RULES - 1);
    chart[(1 * NTOK + s) * RULES + r] = E[tokens[s] * RULES + r];
  }
}

__global__ void cyk_final_kernel(const float* __restrict__ chart,
                                 float* __restrict__ out) {
  if (threadIdx.x == 0) out[0] = chart[(16 * NTOK + 0) * RULES + 0];
}

// ---------------------------------------------------------------------------
extern "C" void kernel_launch(void* const* d_in, const int* in_sizes, int n_in,
                              void* d_out, int out_size, void* d_ws, size_t ws_size,
                              hipStream_t stream) {
  const int*   tokens = (const int*)d_in[0];
  const float* W      = (const float*)d_in[1];
  const float* E      = (const float*)d_in[2];
  float*       chart  = (float*)d_ws;           // 68KB chart in workspace
  float*       out    = (float*)d_out;

  hipLaunchKernelGGL(cyk_init_kernel, dim3(1), dim3(1024), 0, stream,
                     tokens, E, chart);

  for (int L = 2; L <= NTOK; ++L) {
    const int blocks = (NTOK + 1 - L) * (L - 1);   // one block per (s, k)
    hipLaunchKernelGGL(cyk_level_kernel, dim3(blocks), dim3(256), LDS_BYTES, stream,
                       W, chart, L);
  }

  hipLaunchKernelGGL(cyk_final_kernel, dim3(1), dim3(1), 0, stream, chart, out);
}